// GCNLayer_81097572483636
// MI455X (gfx1250) — compile-verified
//
#include <hip/hip_runtime.h>

typedef __attribute__((ext_vector_type(2))) float v2f;
typedef __attribute__((ext_vector_type(8))) float v8f;

#define DIN   256
#define DOUT  64
// LDS holds W as K-pairs: wpair[kk][c] = (W[2kk][c], W[2kk+1][c]), kk in [0,128), c in [0,64).
// Pair-row stride 160 words: 160 mod 64 = 32 -> upper lane half (kk+1) hits banks +32,
// so the per-lane ds_load_b64 B-fragment fetch is bank-conflict-free across the wave.
#define PSTRIDE 160

// ---------------- zero out + degree buffers ----------------
__global__ void k_zero(float* __restrict__ out, int n_out,
                       float* __restrict__ degs, int n_degs) {
    int i = blockIdx.x * blockDim.x + threadIdx.x;
    if (i < n_out)  out[i]  = 0.0f;
    if (i < n_degs) degs[i] = 0.0f;
}

// ---------------- degree build (float atomics; counts are exact in fp32) ----------------
__global__ void k_deg(const int* __restrict__ src, const int* __restrict__ dst,
                      float* __restrict__ outdeg, float* __restrict__ indeg, int E) {
    int e = blockIdx.x * blockDim.x + threadIdx.x;
    if (e < E) {
        atomicAdd(&outdeg[src[e]], 1.0f);
        atomicAdd(&indeg[dst[e]], 1.0f);
    }
}

// ---------------- degrees -> rsqrt norms, in place ----------------
__global__ void k_norm(float* __restrict__ outdeg, float* __restrict__ indeg, int n) {
    int i = blockIdx.x * blockDim.x + threadIdx.x;
    if (i < n) {
        outdeg[i] = rsqrtf(fmaxf(outdeg[i], 1.0f));
        indeg[i]  = rsqrtf(fmaxf(indeg[i], 1.0f));
    }
}

// ---------------- h = (x * norm_src[:,None]) @ W via V_WMMA_F32_16X16X4_F32 ----------------
// Block = 128 threads = 4 waves; each wave computes a 16-row x 64-col tile of h.
__global__ __launch_bounds__(128)
void k_gemm_wmma(const float* __restrict__ x, const float* __restrict__ W,
                 const float* __restrict__ norm_src, float* __restrict__ h, int n) {
    __shared__ float wlds[(DIN / 2) * PSTRIDE];   // 80 KB

    // Stage W (256x64 fp32) as K-pairs: wlds[kk*PSTRIDE + 2c + {0,1}] = W[2kk][c], W[2kk+1][c].
    for (int i = threadIdx.x; i < (DIN / 2) * DOUT; i += blockDim.x) {
        int kk = i >> 6;          // pair index
        int c  = i & 63;          // column
        float w0 = W[(2 * kk + 0) * DOUT + c];
        float w1 = W[(2 * kk + 1) * DOUT + c];
        v2f p; p[0] = w0; p[1] = w1;
        *(v2f*)&wlds[kk * PSTRIDE + 2 * c] = p;
    }
    __syncthreads();

    const int wave = threadIdx.x >> 5;   // 0..3
    const int lane = threadIdx.x & 31;
    const int half = lane >> 4;          // 0 = lanes 0-15, 1 = lanes 16-31
    const int l16  = lane & 15;
    const int koff = half * 2;           // A/B upper lane half holds K+2

    const int rowBase = (blockIdx.x * 4 + wave) * 16;
    const int row  = rowBase + l16;
    const int rowc = (row < n) ? row : (n - 1);         // clamp: keep loads in-bounds
    const float ns = (row < n) ? norm_src[rowc] : 0.0f; // zero contribution for pad rows

    const float* __restrict__ xrow = x + (size_t)rowc * DIN;

    v8f acc0 = {}, acc1 = {}, acc2 = {}, acc3 = {};

    for (int k0 = 0; k0 < DIN; k0 += 4) {
        // A fragment: 16x4 f32. Lane holds row M=l16; VGPR0=K(koff), VGPR1=K(koff+1).
        float2 xa = *(const float2*)(xrow + k0 + koff);
        v2f a; a[0] = xa.x * ns; a[1] = xa.y * ns;

        // B fragments: one aligned 8B LDS load per N-tile lands the (K, K+1) pair
        // directly in an adjacent VGPR pair (no marshalling movs).
        const float* wk = &wlds[((k0 >> 1) + half) * PSTRIDE];
        v2f b0 = *(const v2f*)&wk[2 * (l16 +  0)];
        v2f b1 = *(const v2f*)&wk[2 * (l16 + 16)];
        v2f b2 = *(const v2f*)&wk[2 * (l16 + 32)];
        v2f b3 = *(const v2f*)&wk[2 * (l16 + 48)];

        acc0 = __builtin_amdgcn_wmma_f32_16x16x4_f32(false, a, false, b0, (short)0, acc0, false, false);
        acc1 = __builtin_amdgcn_wmma_f32_16x16x4_f32(false, a, false, b1, (short)0, acc1, false, false);
        acc2 = __builtin_amdgcn_wmma_f32_16x16x4_f32(false, a, false, b2, (short)0, acc2, false, false);
        acc3 = __builtin_amdgcn_wmma_f32_16x16x4_f32(false, a, false, b3, (short)0, acc3, false, false);
    }

    // D layout: VGPR v -> row (v + 8*half), col = l16 (+16 per N-tile).
    const int rowD0 = rowBase + half * 8;
    #pragma unroll
    for (int v = 0; v < 8; ++v) {
        int r = rowD0 + v;
        if (r < n) {
            float* hr = h + (size_t)r * DOUT + l16;
            hr[0]  = acc0[v];
            hr[16] = acc1[v];
            hr[32] = acc2[v];
            hr[48] = acc3[v];
        }
    }
}

// ---------------- scatter-add: out[dst] += h[src], 64 lanes per edge ----------------
__global__ void k_scatter(const float* __restrict__ h, const int* __restrict__ src,
                          const int* __restrict__ dst, float* __restrict__ out, int E) {
    long long idx = (long long)blockIdx.x * blockDim.x + threadIdx.x;
    int e = (int)(idx >> 6);
    if (e < E) {
        int f = (int)(idx & 63);
        atomicAdd(&out[(size_t)dst[e] * DOUT + f], h[(size_t)src[e] * DOUT + f]);
    }
}

// ---------------- epilogue: out = relu(out * norm_dst + b), in place ----------------
__global__ void k_final(float* __restrict__ out, const float* __restrict__ norm_dst,
                        const float* __restrict__ b, int n) {
    long long idx = (long long)blockIdx.x * blockDim.x + threadIdx.x;
    if (idx < (long long)n * DOUT) {
        int i = (int)(idx >> 6);
        int f = (int)(idx & 63);
        float v = fmaf(out[idx], norm_dst[i], b[f]);
        out[idx] = fmaxf(v, 0.0f);
    }
}

extern "C" void kernel_launch(void* const* d_in, const int* in_sizes, int n_in,
                              void* d_out, int out_size, void* d_ws, size_t ws_size,
                              hipStream_t stream) {
    const float* x   = (const float*)d_in[0];
    const float* W   = (const float*)d_in[1];
    const float* b   = (const float*)d_in[2];
    const int*   src = (const int*)d_in[3];
    const int*   dst = (const int*)d_in[4];

    const int n = in_sizes[0] / DIN;   // 50000
    const int E = in_sizes[3];         // 800000

    float* out    = (float*)d_out;
    float* h      = (float*)d_ws;                 // n*64 floats
    float* outdeg = h + (size_t)n * DOUT;         // n floats
    float* indeg  = outdeg + n;                   // n floats  (outdeg/indeg contiguous)

    const int threads = 256;

    // 1) zero accumulator (d_out) and both degree arrays
    {
        int n_out  = n * DOUT;
        int n_degs = 2 * n;
        k_zero<<<(n_out + threads - 1) / threads, threads, 0, stream>>>(out, n_out, outdeg, n_degs);
    }
    // 2) degrees
    k_deg<<<(E + threads - 1) / threads, threads, 0, stream>>>(src, dst, outdeg, indeg, E);
    // 3) norms (in place)
    k_norm<<<(n + threads - 1) / threads, threads, 0, stream>>>(outdeg, indeg, n);
    // 4) WMMA GEMM: h = (x * norm_src) @ W
    {
        int blocks = (n + 63) / 64;   // 64 rows per block (4 waves x 16 rows)
        k_gemm_wmma<<<blocks, 128, 0, stream>>>(x, W, outdeg, h, n);
    }
    // 5) scatter-add into out
    {
        long long work = (long long)E * DOUT;
        int blocks = (int)((work + threads - 1) / threads);
        k_scatter<<<blocks, threads, 0, stream>>>(h, src, dst, out, E);
    }
    // 6) epilogue
    {
        long long work = (long long)n * DOUT;
        int blocks = (int)((work + threads - 1) / threads);
        k_final<<<blocks, threads, 0, stream>>>(out, indeg, b, n);
    }
}